// RouterModel_3281355014339
// MI455X (gfx1250) — compile-verified
//
#include <hip/hip_runtime.h>
#include <stdint.h>

typedef float v2f __attribute__((ext_vector_type(2)));
typedef float v8f __attribute__((ext_vector_type(8)));
typedef float f4  __attribute__((ext_vector_type(4)));

// ---------------------------------------------------------------------------
// Kernel 1: gate GEMM via V_WMMA_F32_16X16X4_F32, one 16-row tile per wave32.
//   A (16x4 f32): lane m=lane&15 holds row row0+m; VGPR0 = K(klo), VGPR1 = K(klo+1)
//                 with klo = 2*(lane>>4)  (ISA 7.12.2 "32-bit A-Matrix 16x4").
//   B (4x16 f32): column n = lane&15; only n<2 are real gate columns, rest 0.
//   C (16x16 f32): row M = v + 8*(lane>=16), col N = lane&15  -> logits at
//                 cols 0,1, i.e. lanes {0,1} (rows v) and {16,17} (rows v+8).
// Emits the per-row route bit (argmax over 2 logits, tie -> 0) to `route`.
// ---------------------------------------------------------------------------
__global__ __launch_bounds__(256) void gate_wmma_kernel(
    const float* __restrict__ x, const float* __restrict__ w,
    unsigned char* __restrict__ route, int d)
{
  const int lane = threadIdx.x & 31;
  const int wid  = threadIdx.x >> 5;
  const int tile = blockIdx.x * (blockDim.x >> 5) + wid;   // 16 rows per tile
  const int row0 = tile << 4;
  const int m    = lane & 15;       // column of B / C, row-within-tile of A
  const int klo  = (lane >> 4) << 1;

  const float* xrow = x + (size_t)(row0 + m) * (size_t)d;

  v8f c0 = {};
  v8f c1 = {};

  // K loop: 4 K-values per WMMA, unrolled x2 with independent accumulators.
  for (int kb = 0; kb < d; kb += 8) {
    v2f a0 = *(const v2f*)(xrow + kb + klo);       // two consecutive K (b64)
    v2f a1 = *(const v2f*)(xrow + kb + 4 + klo);
    v2f b0 = (v2f){0.0f, 0.0f};
    v2f b1 = (v2f){0.0f, 0.0f};
    if (m < 2) {                                   // real gate columns only
      b0.x = w[(kb + klo    ) * 2 + m];
      b0.y = w[(kb + klo + 1) * 2 + m];
      b1.x = w[(kb + 4 + klo    ) * 2 + m];
      b1.y = w[(kb + 4 + klo + 1) * 2 + m];
    }
    c0 = __builtin_amdgcn_wmma_f32_16x16x4_f32(false, a0, false, b0,
                                               (short)0, c0, false, false);
    c1 = __builtin_amdgcn_wmma_f32_16x16x4_f32(false, a1, false, b1,
                                               (short)0, c1, false, false);
  }
  v8f c = c0 + c1;

  // Gather the two logit columns per row via v_readlane, build a 16-bit mask.
  unsigned int mask = 0;
#pragma unroll
  for (int v = 0; v < 8; ++v) {
    int i0 = __builtin_amdgcn_readlane(__float_as_int(c[v]), 0);   // row v,   col 0
    int i1 = __builtin_amdgcn_readlane(__float_as_int(c[v]), 1);   // row v,   col 1
    int i2 = __builtin_amdgcn_readlane(__float_as_int(c[v]), 16);  // row v+8, col 0
    int i3 = __builtin_amdgcn_readlane(__float_as_int(c[v]), 17);  // row v+8, col 1
    if (__int_as_float(i1) > __int_as_float(i0)) mask |= 1u << v;
    if (__int_as_float(i3) > __int_as_float(i2)) mask |= 1u << (v + 8);
  }
  if (lane < 16)
    route[row0 + lane] = (unsigned char)((mask >> lane) & 1u);
}

// ---------------------------------------------------------------------------
// Kernel 2: streaming scatter/gather. x is L2-resident after kernel 1
// (134 MB < 192 MB L2), so the reload is cheap; outputs are written with
// non-temporal b128 stores since they are never re-read.
// ---------------------------------------------------------------------------
__global__ __launch_bounds__(256) void scatter_gather_kernel(
    const float* __restrict__ x, const unsigned char* __restrict__ route,
    float* __restrict__ x0, float* __restrict__ x1, float* __restrict__ out,
    int dq /* = d/4 */)
{
  const size_t i   = (size_t)blockIdx.x * blockDim.x + threadIdx.x; // float4 idx
  const size_t row = i / (size_t)dq;

  const f4 v = ((const f4*)x)[i];
  const unsigned char r = route[row];
  const f4 z = (f4){0.0f, 0.0f, 0.0f, 0.0f};
  const f4 a = r ? z : v;   // route==0 stream
  const f4 b = r ? v : z;   // route==1 stream

  __builtin_nontemporal_store(a, ((f4*)x0)  + i);
  __builtin_nontemporal_store(b, ((f4*)x1)  + i);
  __builtin_nontemporal_store(v, ((f4*)out) + i);  // x0 + x1 == x exactly
}

extern "C" void kernel_launch(void* const* d_in, const int* in_sizes, int n_in,
                              void* d_out, int out_size, void* d_ws, size_t ws_size,
                              hipStream_t stream) {
  const float* x = (const float*)d_in[0];
  const float* w = (const float*)d_in[1];

  const int d = in_sizes[1] / 2;            // 2048
  const int n = in_sizes[0] / d;            // 16384
  const size_t nd = (size_t)n * (size_t)d;

  float* x0  = (float*)d_out;
  float* x1  = (float*)d_out + nd;
  float* out = (float*)d_out + 2 * nd;
  unsigned char* route = (unsigned char*)d_ws;

  // Gate: n/16 tiles, one per wave, 8 waves (256 threads) per block.
  const int tiles  = n >> 4;                               // 1024
  const int blocks = (tiles + 7) / 8;                      // 128
  gate_wmma_kernel<<<blocks, 256, 0, stream>>>(x, w, route, d);

  // Streaming pass: one float4 per thread.
  const size_t total4 = nd >> 2;                           // 8,388,608
  const int blocks2 = (int)((total4 + 255) / 256);         // 32768
  scatter_gather_kernel<<<blocks2, 256, 0, stream>>>(x, route, x0, x1, out, d >> 2);
}